// TopKActivation_64647847740068
// MI455X (gfx1250) — compile-verified
//
#include <hip/hip_runtime.h>
#include <stdint.h>

// Problem shape from the reference: x is [8192, 16384] fp32, K = 32.
#define COLS 16384
#define TPB  512
#define KSEL 32u
#define CHUNKS 8   // 8 chunks * 512 threads * 4 floats = 16384 floats per row

// Order-preserving map fp32 -> uint32 (larger float => larger uint).
__device__ __forceinline__ uint32_t f2ord(float f) {
  uint32_t u = __float_as_uint(f);
  return u ^ (uint32_t)(((int32_t)u >> 31) | (int32_t)0x80000000);
}

__global__ __launch_bounds__(TPB) void topk_act_kernel(const float* __restrict__ x,
                                                       float* __restrict__ out) {
  __shared__ float smem[COLS];                     // staged row (64 KB)
  __shared__ __align__(16) uint32_t hist[256];     // radix histogram
  __shared__ uint32_t s_prefix;
  __shared__ uint32_t s_kremain;
  __shared__ uint32_t s_tie;

  const uint32_t t   = threadIdx.x;
  const uint32_t row = blockIdx.x;
  const float* rowp  = x + (size_t)row * COLS;

  // LDS byte address of smem within this workgroup's allocation:
  // generic pointers to LDS carry the allocation-relative offset in addr[31:0].
  const uint32_t lds_base = (uint32_t)(uintptr_t)(void*)smem;

  // ---- Stage the whole row into LDS with async global->LDS B128 copies ----
  // GVS addressing: mem = SADDR(64b) + VADDR(32b byte offset); VDST = LDS byte addr.
#pragma unroll
  for (int i = 0; i < CHUNKS; ++i) {
    uint32_t goff  = (uint32_t)(i * TPB + t) * 16u;   // byte offset of 16B chunk
    uint32_t laddr = lds_base + goff;
    asm volatile("global_load_async_to_lds_b128 %0, %1, %2"
                 :: "v"(laddr), "v"(goff), "s"(rowp)
                 : "memory");
  }
  asm volatile("s_wait_asynccnt 0" ::: "memory");
  __syncthreads();

  // ---- Pull this thread's 32 values into registers (ds_load_b128) ----
  float4 vals[CHUNKS];
#pragma unroll
  for (int i = 0; i < CHUNKS; ++i) {
    vals[i] = *reinterpret_cast<const float4*>(&smem[(size_t)(i * TPB + t) * 4]);
  }

  // ---- Exact radix select: find ordered-uint of the K-th largest value ----
  const int      shifts[4]  = {24, 16, 8, 0};
  const uint32_t himasks[4] = {0x00000000u, 0xFF000000u, 0xFFFF0000u, 0xFFFFFF00u};

  uint32_t prefix  = 0;
  uint32_t kremain = KSEL;

#pragma unroll
  for (int round = 0; round < 4; ++round) {
    const int      shift  = shifts[round];
    const uint32_t himask = himasks[round];

    if (t < 256) hist[t] = 0;
    __syncthreads();

#pragma unroll
    for (int i = 0; i < CHUNKS; ++i) {
      const float* f = reinterpret_cast<const float*>(&vals[i]);
#pragma unroll
      for (int j = 0; j < 4; ++j) {
        uint32_t u = f2ord(f[j]);
        if ((u & himask) == prefix) {
          atomicAdd(&hist[(u >> shift) & 0xFFu], 1u);
        }
      }
    }
    __syncthreads();

    // Wave-0 parallel descending scan: lane i owns bins [8i, 8i+8).
    if (t < 32) {
      uint32_t h[8];
      {
        const uint4* hp = reinterpret_cast<const uint4*>(&hist[t * 8]);
        uint4 a = hp[0], b = hp[1];
        h[0] = a.x; h[1] = a.y; h[2] = a.z; h[3] = a.w;
        h[4] = b.x; h[5] = b.y; h[6] = b.z; h[7] = b.w;
      }
      uint32_t laneSum = 0;
#pragma unroll
      for (int j = 0; j < 8; ++j) laneSum += h[j];

      // Inclusive suffix-sum across lanes: suff[i] = sum_{l>=i} laneSum[l]
      uint32_t suff = laneSum;
#pragma unroll
      for (int off = 1; off < 32; off <<= 1) {
        uint32_t v = __shfl_down(suff, off, 32);
        if ((int)t + off < 32) suff += v;
      }
      uint32_t suffNext = __shfl_down(suff, 1, 32);
      if (t == 31) suffNext = 0;

      // Exactly one lane holds the chunk where the cumulative count crosses kremain.
      if (suff >= kremain && suffNext < kremain) {
        uint32_t acc = suffNext;
#pragma unroll
        for (int b = 7; b >= 0; --b) {
          uint32_t c = h[b];
          if (acc + c >= kremain) {
            s_prefix  = prefix | ((t * 8u + (uint32_t)b) << shift);
            s_kremain = kremain - acc;   // part of K still inside this bin
            break;
          }
          acc += c;
        }
      }
    }
    __syncthreads();
    prefix  = s_prefix;
    kremain = s_kremain;
  }

  const uint32_t thr = prefix;     // ordered-uint of the 32nd-largest value
  // kremain = number of elements equal to thr that belong to the top-K (ties)

  if (t == 0) s_tie = 0;
  __syncthreads();

  // ---- Output pass: relu(top-K) scattered in place, zeros elsewhere ----
  float* orow = out + (size_t)row * COLS;
#pragma unroll
  for (int i = 0; i < CHUNKS; ++i) {
    const float* f = reinterpret_cast<const float*>(&vals[i]);
    float4 o;
    float* op = reinterpret_cast<float*>(&o);
#pragma unroll
    for (int j = 0; j < 4; ++j) {
      float    v = f[j];
      uint32_t u = f2ord(v);
      float    r = 0.0f;
      if (u > thr) {
        r = fmaxf(v, 0.0f);
      } else if (u == thr) {
        if (atomicAdd(&s_tie, 1u) < kremain) r = fmaxf(v, 0.0f);
      }
      op[j] = r;
    }
    *reinterpret_cast<float4*>(&orow[(size_t)(i * TPB + t) * 4]) = o;
  }
}

extern "C" void kernel_launch(void* const* d_in, const int* in_sizes, int n_in,
                              void* d_out, int out_size, void* d_ws, size_t ws_size,
                              hipStream_t stream) {
  const float* x = (const float*)d_in[0];
  float* out     = (float*)d_out;
  const int rows = in_sizes[0] / COLS;   // 8192 for the reference shape
  hipLaunchKernelGGL(topk_act_kernel, dim3(rows), dim3(TPB), 0, stream, x, out);
}